// SS2D_50826642981197
// MI455X (gfx1250) — compile-verified
//
#include <hip/hip_runtime.h>
#include <math.h>

// ---------------- problem constants (from reference) ----------------
constexpr int BN   = 2;            // batch
constexpr int HS   = 56, WS = 56;  // spatial
constexpr int LS   = HS * WS;      // 3136
constexpr int CM   = 96;           // d_model
constexpr int DI   = 192;          // d_inner
constexpr int NS   = 16;           // d_state
constexpr int BL   = BN * LS;      // 6272 rows
constexpr int NPROJ = 2 * DI;      // 384
constexpr int SSMC  = 4 * (2 * NS + 1);  // 132
constexpr int SSMP  = 144;               // padded to multiple of 16

typedef float v2f __attribute__((ext_vector_type(2)));
typedef float v8f __attribute__((ext_vector_type(8)));

__device__ __forceinline__ float silu_f(float v) {
    return v / (1.f + __expf(-v) * 0.f + expf(-v) * 1.f); // keep precise expf
}

// ---------------- LayerNorm over last dim (96) ----------------
__global__ __launch_bounds__(CM) void ln_kernel(const float* __restrict__ x,
                                                const float* __restrict__ g,
                                                const float* __restrict__ be,
                                                float* __restrict__ xn) {
    const int row = blockIdx.x;          // 0..BL-1
    const int c   = threadIdx.x;         // 0..95
    __shared__ float sv[CM];
    float v = x[row * CM + c];
    sv[c] = v;
    __syncthreads();
    float sum = 0.f;
    #pragma unroll 8
    for (int j = 0; j < CM; ++j) sum += sv[j];
    const float mu = sum * (1.f / CM);
    float var = 0.f;
    #pragma unroll 8
    for (int j = 0; j < CM; ++j) { float d = sv[j] - mu; var += d * d; }
    var *= (1.f / CM);
    xn[row * CM + c] = (v - mu) * rsqrtf(var + 1e-5f) * g[c] + be[c];
}

// ---------------- pad x_proj_w (192x132 -> 192x144) ----------------
__global__ void padw_kernel(const float* __restrict__ w, float* __restrict__ pw) {
    int idx = blockIdx.x * 256 + threadIdx.x;
    if (idx < DI * SSMP) {
        int k = idx / SSMP, n = idx % SSMP;
        pw[idx] = (n < SSMC) ? w[k * SSMC + n] : 0.f;
    }
}

// ---------------- fp32 WMMA 16x16 tile GEMM core (one wave / tile) ----------------
// A: MxK row-major, B: KxN row-major. K multiple of 4.
__device__ __forceinline__ v8f wmma_tile_f32(const float* __restrict__ A, int lda,
                                             const float* __restrict__ B, int ldb,
                                             int K, int m0, int n0) {
    const int ln = threadIdx.x & 31;
    const int mr = ln & 15;       // M (for A) / N (for B/C) within tile
    const int hi = ln >> 4;       // lane-group selector
    v8f acc = {};
    for (int k0 = 0; k0 < K; k0 += 4) {
        v2f a, b;
        // A 16x4 layout: elem r -> K = k0 + r + 2*hi, row = m0+mr
        const float* arow = A + (size_t)(m0 + mr) * lda + k0 + 2 * hi;
        a[0] = arow[0];
        a[1] = arow[1];
        // B 4x16 layout: elem r -> K = k0 + r + 2*hi, col = n0+mr
        b[0] = B[(size_t)(k0 + 2 * hi + 0) * ldb + n0 + mr];
        b[1] = B[(size_t)(k0 + 2 * hi + 1) * ldb + n0 + mr];
        acc = __builtin_amdgcn_wmma_f32_16x16x4_f32(false, a, false, b,
                                                    (short)0, acc, false, false);
    }
    return acc;
}

// ---------------- stage 2: xz = xn @ in_proj_w ; split -> xproj, silu(z) ----------------
__global__ __launch_bounds__(32) void gemm_inproj(const float* __restrict__ xn,
                                                  const float* __restrict__ w,
                                                  float* __restrict__ xproj,
                                                  float* __restrict__ siluz) {
    const int m0 = blockIdx.x * 16;
    const int n0 = blockIdx.y * 16;
    v8f acc = wmma_tile_f32(xn, CM, w, NPROJ, CM, m0, n0);
    const int ln = threadIdx.x & 31;
    const int mr = ln & 15, hi = ln >> 4;
    #pragma unroll
    for (int r = 0; r < 8; ++r) {
        int row = m0 + r + 8 * hi;
        int col = n0 + mr;
        float v = acc[r];
        if (col < DI) {
            xproj[(size_t)row * DI + col] = v;
        } else {
            siluz[(size_t)row * DI + (col - DI)] = v / (1.f + expf(-v));
        }
    }
}

// ---------------- stage 3: depthwise 3x3 conv + bias + SiLU ----------------
__global__ __launch_bounds__(DI) void conv_kernel(const float* __restrict__ xproj,
                                                  const float* __restrict__ cw,
                                                  const float* __restrict__ cb,
                                                  float* __restrict__ xconv) {
    const int bl = blockIdx.x;     // b*LS + h*WS + w
    const int d  = threadIdx.x;    // channel
    const int b  = bl / LS;
    const int l  = bl - b * LS;
    const int hh = l / WS;
    const int ww = l - hh * WS;
    float acc = cb[d];
    #pragma unroll
    for (int i = 0; i < 3; ++i) {
        int yy = hh + i - 1;
        if (yy < 0 || yy >= HS) continue;
        #pragma unroll
        for (int j = 0; j < 3; ++j) {
            int xx = ww + j - 1;
            if (xx < 0 || xx >= WS) continue;
            acc += cw[d * 9 + i * 3 + j] *
                   xproj[((size_t)b * LS + yy * WS + xx) * DI + d];
        }
    }
    xconv[(size_t)bl * DI + d] = acc / (1.f + expf(-acc));
}

// ---------------- stage 4: ssm = x1 @ x_proj_w_padded (6272x192 @ 192x144) ----------------
__global__ __launch_bounds__(32) void gemm_xproj(const float* __restrict__ xconv,
                                                 const float* __restrict__ pw,
                                                 float* __restrict__ ssm) {
    const int m0 = blockIdx.x * 16;
    const int n0 = blockIdx.y * 16;
    v8f acc = wmma_tile_f32(xconv, DI, pw, SSMP, DI, m0, n0);
    const int ln = threadIdx.x & 31;
    const int mr = ln & 15, hi = ln >> 4;
    #pragma unroll
    for (int r = 0; r < 8; ++r) {
        int row = m0 + r + 8 * hi;
        int col = n0 + mr;
        ssm[(size_t)row * SSMP + col] = acc[r];
    }
}

// ---------------- stage 5: 4-direction selective scan ----------------
// grid.x = 8 (k = bx>>1, b = bx&1); 192 threads, one channel each.
__global__ __launch_bounds__(DI) void scan_kernel(const float* __restrict__ ssm,
                                                  const float* __restrict__ xconv,
                                                  const float* __restrict__ dt_w,
                                                  const float* __restrict__ dt_b,
                                                  const float* __restrict__ A_log,
                                                  const float* __restrict__ Ds,
                                                  float* __restrict__ ys) {
    const int k = blockIdx.x >> 1;
    const int b = blockIdx.x & 1;
    const int d = threadIdx.x;
    __shared__ float sBC[2 * NS + 1];   // delta0, B[16], C[16] (uniform per step)

    float Arow[NS];
    #pragma unroll
    for (int n = 0; n < NS; ++n) Arow[n] = -expf(A_log[d * NS + n]);
    const float dw  = dt_w[k * DI + d];
    const float db  = dt_b[k * DI + d];
    const float dsv = Ds[k * DI + d];

    float h[NS];
    #pragma unroll
    for (int n = 0; n < NS; ++n) h[n] = 0.f;

    for (int t = 0; t < LS; ++t) {
        __syncthreads();
        if (d < 2 * NS + 1)
            sBC[d] = ssm[((size_t)b * LS + t) * SSMP + k * (2 * NS + 1) + d];
        __syncthreads();

        // delta = softplus(delta0 * dt_w + dt_b)
        const float spi   = sBC[0] * dw + db;
        const float delta = fmaxf(spi, 0.f) + log1pf(expf(-fabsf(spi)));

        // input position for this direction (ssm row is always row-major t)
        int l;
        if (k == 0)      l = t;
        else if (k == 1) l = LS - 1 - t;
        else {
            int tt = (k == 2) ? t : (LS - 1 - t);
            int w2 = tt / HS, h2 = tt - w2 * HS;   // tt = w*H + h
            l = h2 * WS + w2;
        }
        const float xt = xconv[((size_t)b * LS + l) * DI + d];
        const float dx = delta * xt;

        float y = 0.f;
        #pragma unroll
        for (int n = 0; n < NS; ++n) {
            float dA = expf(delta * Arow[n]);
            h[n] = dA * h[n] + sBC[1 + n] * dx;
            y += h[n] * sBC[1 + NS + n];
        }
        ys[(((size_t)k * BN + b) * LS + t) * DI + d] = y + xt * dsv;
    }
}

// ---------------- stage 6: merge directions, multiply by silu(z) ----------------
__global__ __launch_bounds__(DI) void combine_kernel(const float* __restrict__ ys,
                                                     const float* __restrict__ siluz,
                                                     float* __restrict__ yz) {
    const int bl = blockIdx.x;
    const int d  = threadIdx.x;
    const int b  = bl / LS;
    const int l  = bl - b * LS;
    const int hh = l / WS;
    const int ww = l - hh * WS;
    const int t2 = ww * HS + hh;          // transposed scan index
    float y = ys[(((size_t)0 * BN + b) * LS + l) * DI + d]
            + ys[(((size_t)1 * BN + b) * LS + (LS - 1 - l)) * DI + d]
            + ys[(((size_t)2 * BN + b) * LS + t2) * DI + d]
            + ys[(((size_t)3 * BN + b) * LS + (LS - 1 - t2)) * DI + d];
    yz[(size_t)bl * DI + d] = y * siluz[(size_t)bl * DI + d];
}

// ---------------- stage 7: out = yz @ out_proj_w + resid ----------------
__global__ __launch_bounds__(32) void gemm_outproj(const float* __restrict__ yz,
                                                   const float* __restrict__ w,
                                                   const float* __restrict__ resid,
                                                   float* __restrict__ out) {
    const int m0 = blockIdx.x * 16;
    const int n0 = blockIdx.y * 16;
    v8f acc = wmma_tile_f32(yz, DI, w, CM, DI, m0, n0);
    const int ln = threadIdx.x & 31;
    const int mr = ln & 15, hi = ln >> 4;
    #pragma unroll
    for (int r = 0; r < 8; ++r) {
        int row = m0 + r + 8 * hi;
        int col = n0 + mr;
        out[(size_t)row * CM + col] = acc[r] + resid[(size_t)row * CM + col];
    }
}

// ---------------- host launcher ----------------
extern "C" void kernel_launch(void* const* d_in, const int* in_sizes, int n_in,
                              void* d_out, int out_size, void* d_ws, size_t ws_size,
                              hipStream_t stream) {
    const float* x        = (const float*)d_in[0];   // (B,H,W,96)
    const float* ln_g     = (const float*)d_in[1];
    const float* ln_b     = (const float*)d_in[2];
    const float* in_pw    = (const float*)d_in[3];   // (96,384)
    const float* conv_w   = (const float*)d_in[4];   // (192,1,3,3)
    const float* conv_b   = (const float*)d_in[5];
    const float* x_pw     = (const float*)d_in[6];   // (192,132)
    const float* dt_w     = (const float*)d_in[7];   // (4,192)
    const float* dt_b     = (const float*)d_in[8];
    const float* A_log    = (const float*)d_in[9];   // (192,16)
    const float* Ds       = (const float*)d_in[10];  // (4,192)
    const float* out_pw   = (const float*)d_in[11];  // (192,96)
    float*       out      = (float*)d_out;

    // workspace layout (floats)
    float* ws    = (float*)d_ws;
    float* xn    = ws;                         // BL*96
    float* xproj = xn    + (size_t)BL * CM;    // BL*192
    float* siluz = xproj + (size_t)BL * DI;    // BL*192
    float* xconv = siluz + (size_t)BL * DI;    // BL*192
    float* padw  = xconv + (size_t)BL * DI;    // 192*144
    float* ssm   = padw  + (size_t)DI * SSMP;  // BL*144
    float* ysb   = ssm   + (size_t)BL * SSMP;  // 4*B*L*192
    float* yz    = ysb   + (size_t)4 * BN * LS * DI; // BL*192

    // 1. LayerNorm
    ln_kernel<<<BL, CM, 0, stream>>>(x, ln_g, ln_b, xn);
    // 2. pad x_proj_w
    padw_kernel<<<(DI * SSMP + 255) / 256, 256, 0, stream>>>(x_pw, padw);
    // 3. in_proj GEMM (WMMA f32): 6272x96 @ 96x384
    gemm_inproj<<<dim3(BL / 16, NPROJ / 16), 32, 0, stream>>>(xn, in_pw, xproj, siluz);
    // 4. depthwise conv + SiLU
    conv_kernel<<<BL, DI, 0, stream>>>(xproj, conv_w, conv_b, xconv);
    // 5. x_proj GEMM (WMMA f32): 6272x192 @ 192x144
    gemm_xproj<<<dim3(BL / 16, SSMP / 16), 32, 0, stream>>>(xconv, padw, ssm);
    // 6. selective scan: 8 blocks (4 dirs x 2 batch)
    scan_kernel<<<4 * BN, DI, 0, stream>>>(ssm, xconv, dt_w, dt_b, A_log, Ds, ysb);
    // 7. merge + silu(z) gate
    combine_kernel<<<BL, DI, 0, stream>>>(ysb, siluz, yz);
    // 8. out_proj GEMM + residual: 6272x192 @ 192x96
    gemm_outproj<<<dim3(BL / 16, CM / 16), 32, 0, stream>>>(yz, out_pw, x, out);
}